// TcSpace_15814069583885
// MI455X (gfx1250) — compile-verified
//
#include <hip/hip_runtime.h>
#include <hip/hip_bf16.h>
#include <math.h>

// ---------------------------------------------------------------------------
// Problem constants (from reference)
// ---------------------------------------------------------------------------
#define TT 4
#define BB 512
#define GG 16
#define NN 256        // GG*GG
#define DD 64
#define FRAME (NN * DD)       // 16384 floats per (t,b) frame
#define CSTR 68               // LDS stride per cell: 272B = 17*16B (aligned, breaks 256B bank period)
#define NPAIR ((TT - 1) * BB) // 1536 frame-pair blocks

typedef float v2f __attribute__((ext_vector_type(2)));
typedef float v8f __attribute__((ext_vector_type(8)));

// Workspace layout (floats):
//  [0      .. 1535 ]  per-pair-block partial: z_what_loss + pool_loss
//  [1536   .. 3071 ]  per-pair-block partial: obj_loss (unscaled)
//  [3072   .. 3327 ]  per-block partial: z_pres_loss (256 blocks)
#define WS_OBJ   NPAIR
#define WS_ZPRES (2 * NPAIR)
#define ZPRES_BLOCKS 256

// Dynamic LDS layout (floats)
#define L_SA   0
#define L_SB   (L_SA + NN * CSTR)
#define L_ZPA  (L_SB + NN * CSTR)
#define L_ZPB  (L_ZPA + NN)
#define L_NA   (L_ZPB + NN)
#define L_NB   (L_NA + NN)
#define L_SS   (L_NB + NN)
#define L_RED  (L_SS + GG * 3 * 256)
#define L_TOT  (L_RED + 256)
#define SMEM_BYTES (L_TOT * sizeof(float))

// ---------------------------------------------------------------------------
// Kernel A: one workgroup per (t, b) frame pair.
//   partial0 = sum (zw[t+1]-zw[t])^2  +  pool_loss contribution
//   partial1 = obj_loss contribution (unscaled)
// Obj-loss row-dot matrices via V_WMMA_F32_16X16X4_F32; all bulk data
// movement uses b128 (float4) global/LDS accesses.
// ---------------------------------------------------------------------------
__global__ __launch_bounds__(256) void pair_kernel(const float* __restrict__ zw,
                                                   const float* __restrict__ zpp,
                                                   float* __restrict__ ws) {
    extern __shared__ float smem[];
    float* sA  = smem + L_SA;
    float* sB  = smem + L_SB;
    float* zpA = smem + L_ZPA;
    float* zpB = smem + L_ZPB;
    float* nA  = smem + L_NA;
    float* nB  = smem + L_NB;
    float* sS  = smem + L_SS;
    float* red = smem + L_RED;

    const int p   = blockIdx.x;        // 0 .. NPAIR-1
    const int t   = p >> 9;            // 0..2
    const int b   = p & (BB - 1);
    const int tid = threadIdx.x;       // 0..255

    const float4* A4 = (const float4*)(zw + (size_t)(t * BB + b) * FRAME);
    const float4* B4 = (const float4*)(zw + (size_t)((t + 1) * BB + b) * FRAME);

    // presence probabilities for both frames
    zpA[tid] = zpp[(size_t)(t * BB + b) * NN + tid];
    zpB[tid] = zpp[(size_t)((t + 1) * BB + b) * NN + tid];

    // ---- load both raw frames to LDS (b128), accumulate z_what_loss ----
    float zwl = 0.0f;
    for (int q = tid; q < FRAME / 4; q += 256) {     // 16 iterations
        float4 a  = A4[q];
        float4 bb = B4[q];
        int cell = q >> 4;           // 16 float4 per cell
        int dq   = q & 15;
        ((float4*)(sA + cell * CSTR))[dq] = a;       // 272B cell stride: 16B aligned
        ((float4*)(sB + cell * CSTR))[dq] = bb;
        float d0 = bb.x - a.x, d1 = bb.y - a.y, d2 = bb.z - a.z, d3 = bb.w - a.w;
        zwl += d0 * d0 + d1 * d1 + d2 * d2 + d3 * d3;
    }
    __syncthreads();

    // ---- per-cell L2 norms of raw vectors (b128 LDS reads) ----
    {
        const float4* ra = (const float4*)(sA + tid * CSTR);
        const float4* rb = (const float4*)(sB + tid * CSTR);
        float sa = 0.0f, sb = 0.0f;
        #pragma unroll
        for (int k = 0; k < DD / 4; ++k) {
            float4 x = ra[k];
            sa += x.x * x.x + x.y * x.y + x.z * x.z + x.w * x.w;
            float4 y = rb[k];
            sb += y.x * y.x + y.y * y.y + y.z * y.z + y.w * y.w;
        }
        nA[tid] = sqrtf(sa);
        nB[tid] = sqrtf(sb);
    }
    __syncthreads();

    // ---- pool loss: 3x3 clamped max over ww_t, cosine vs ww_{t+1} ----
    // 4 channels per step: one b128 LDS read per neighbor per 4 channels.
    float poolp;
    {
        const int n = tid, i = n >> 4, j = n & 15;
        const int i0 = (i > 0) ? i - 1 : 0, i1 = (i < 15) ? i + 1 : 15;
        const int j0 = (j > 0) ? j - 1 : 0, j1 = (j < 15) ? j + 1 : 15;
        const float zb = zpB[n];

        // hoist neighbor zp and base pointers (<= 9 neighbors)
        int   nnb = 0;
        const float4* nbp[9];
        float nbw[9];
        for (int ii = i0; ii <= i1; ++ii)
            for (int jj = j0; jj <= j1; ++jj) {
                nbp[nnb] = (const float4*)(sA + (ii * GG + jj) * CSTR);
                nbw[nnb] = zpA[ii * GG + jj];
                ++nnb;
            }

        const float4* rb = (const float4*)(sB + n * CSTR);
        float dot = 0.0f, ns = 0.0f;
        for (int k = 0; k < DD / 4; ++k) {
            float m0 = 0.0f, m1 = 0.0f, m2 = 0.0f, m3 = 0.0f; // ww>=0, center valid
            for (int q = 0; q < nnb; ++q) {
                float4 v = nbp[q][k];
                float w = nbw[q];
                m0 = fmaxf(m0, fabsf(v.x) * w);
                m1 = fmaxf(m1, fabsf(v.y) * w);
                m2 = fmaxf(m2, fabsf(v.z) * w);
                m3 = fmaxf(m3, fabsf(v.w) * w);
            }
            float4 y = rb[k];
            dot += m0 * (fabsf(y.x) * zb) + m1 * (fabsf(y.y) * zb)
                 + m2 * (fabsf(y.z) * zb) + m3 * (fabsf(y.w) * zb);
            ns  += m0 * m0 + m1 * m1 + m2 * m2 + m3 * m3;
        }
        float denom = fmaxf(sqrtf(ns), 1e-6f) * fmaxf(zb * nB[n], 1e-6f);
        float cosv  = dot / denom;
        poolp = -cosv * 0.5f * (zpA[n] + zpB[n]);
    }

    // ---- obj loss, stage 1: WMMA row-dot tiles ----
    // wave w owns prior rows 2w, 2w+1. For each, S_dr = P_i (16x64) x N_{(i+dr-1)&15}^T
    // via 16 chained V_WMMA_F32_16X16X4_F32 (K=64).
    {
        const int wave = tid >> 5;
        const int lane = tid & 31;
        const int la   = lane & 15;          // M index (A) / N index (B)
        const int kh   = (lane >> 4) << 1;   // K sub-offset: 0 or 2

        for (int rr = 0; rr < 2; ++rr) {
            const int i = wave * 2 + rr;                 // prior grid row
            const float* pA = sA + (i * GG + la) * CSTR; // A[m=la][k]
            for (int dr = 0; dr < 3; ++dr) {
                const int r = (i + dr - 1) & 15;         // next grid row (wrap)
                const float* pB = sB + (r * GG + la) * CSTR; // B[k][n=la] = Q[la][k]
                v8f c = {};
                #pragma unroll
                for (int k0 = 0; k0 < DD; k0 += 4) {
                    v2f a, bfr;
                    a.x   = pA[k0 + kh];
                    a.y   = pA[k0 + kh + 1];
                    bfr.x = pB[k0 + kh];
                    bfr.y = pB[k0 + kh + 1];
                    c = __builtin_amdgcn_wmma_f32_16x16x4_f32(
                            false, a, false, bfr, (short)0, c, false, false);
                }
                // C layout: VGPR v, lane L -> row (v + 8*(L>=16)), col (L&15)
                float* dst = sS + (i * 3 + dr) * 256;
                const int rbase = (lane >> 4) * 8;
                #pragma unroll
                for (int v = 0; v < 8; ++v)
                    dst[(rbase + v) * GG + la] = c[v];
            }
        }
    }
    __syncthreads();

    // ---- obj loss, stage 2: per-cell masked sum / max of cosines ----
    float objp = 0.0f;
    {
        const int n = tid, i = n >> 4, j = n & 15;
        if (zpA[n] > 0.5f) {                       // cell_mask (frame t)
            const float inp = 1.0f / fmaxf(nA[n], 1e-8f);
            float s = 0.0f, mx = -1e30f;
            bool any = false;
            for (int dr = 0; dr < 3; ++dr) {
                const int rw = (i + dr - 1) & 15;
                const float* Srow = sS + (i * 3 + dr) * 256 + j * GG;
                for (int dj = 0; dj < 3; ++dj) {
                    const int cl = (j + dj - 1) & 15;
                    const int nb = rw * GG + cl;
                    if (zpB[nb] > 0.5f) {          // nb_mask (frame t+1, rolled)
                        float cosv = Srow[cl] * inp / fmaxf(nB[nb], 1e-8f);
                        s += cosv;
                        mx = fmaxf(mx, cosv);
                        any = true;
                    }
                }
            }
            if (any) objp = s - 5.0f * mx;         // Z_COS_MATCH_WEIGHT = 5
        }
    }

    // ---- deterministic block reductions ----
    red[tid] = zwl + poolp;
    __syncthreads();
    for (int s = 128; s > 0; s >>= 1) {
        if (tid < s) red[tid] += red[tid + s];
        __syncthreads();
    }
    if (tid == 0) ws[p] = red[0];
    __syncthreads();

    red[tid] = objp;
    __syncthreads();
    for (int s = 128; s > 0; s >>= 1) {
        if (tid < s) red[tid] += red[tid + s];
        __syncthreads();
    }
    if (tid == 0) ws[WS_OBJ + p] = red[0];
}

// ---------------------------------------------------------------------------
// Kernel B: z_pres temporal loss.  sim*(deltas) over k=0,1 windows.
// ---------------------------------------------------------------------------
__global__ __launch_bounds__(256) void zpres_kernel(const float* __restrict__ zp,
                                                    float* __restrict__ ws) {
    __shared__ float red[256];
    const int tid    = threadIdx.x;
    const int gid    = blockIdx.x * 256 + tid;
    const int stride = gridDim.x * 256;
    const int FR     = BB * NN;      // 131072 per frame
    float acc = 0.0f;
    for (int idx = gid; idx < 2 * FR; idx += stride) {
        int k = idx >> 17;           // 0 or 1
        int m = idx & (FR - 1);
        float a = zp[(size_t)k * FR + m];
        float b = zp[(size_t)(k + 1) * FR + m];
        float c = zp[(size_t)(k + 2) * FR + m];
        float d1  = c - a;
        float sim = 1.0f - d1 * d1;
        float e1  = c - b;
        float e2  = a - b;
        acc += sim * (e1 * e1 + e2 * e2);
    }
    red[tid] = acc;
    __syncthreads();
    for (int s = 128; s > 0; s >>= 1) {
        if (tid < s) red[tid] += red[tid + s];
        __syncthreads();
    }
    if (tid == 0) ws[WS_ZPRES + blockIdx.x] = red[0];
}

// ---------------------------------------------------------------------------
// Kernel C: deterministic final reduction + scaling.
// ---------------------------------------------------------------------------
__global__ __launch_bounds__(256) void final_kernel(const float* __restrict__ ws,
                                                    const int* __restrict__ gstep,
                                                    float* __restrict__ out) {
    __shared__ float red[256];
    const int tid = threadIdx.x;

    float a = 0.0f, o = 0.0f;
    for (int i = tid; i < NPAIR; i += 256) {
        a += ws[i];
        o += ws[WS_OBJ + i];
    }
    float z = ws[WS_ZPRES + tid];

    red[tid] = a; __syncthreads();
    for (int s = 128; s > 0; s >>= 1) { if (tid < s) red[tid] += red[tid + s]; __syncthreads(); }
    float sumA = red[0]; __syncthreads();

    red[tid] = o; __syncthreads();
    for (int s = 128; s > 0; s >>= 1) { if (tid < s) red[tid] += red[tid + s]; __syncthreads(); }
    float sumO = red[0]; __syncthreads();

    red[tid] = z; __syncthreads();
    for (int s = 128; s > 0; s >>= 1) { if (tid < s) red[tid] += red[tid + s]; __syncthreads(); }

    if (tid == 0) {
        float sumZ    = red[0];
        float scaling = fminf(1.0f, (float)(*gstep) / 20000.0f); // FULL_OBJ_WEIGHT
        out[0] = sumA + sumZ + sumO * scaling * 10.0f;            // OBJ_W = 10
    }
}

// ---------------------------------------------------------------------------
extern "C" void kernel_launch(void* const* d_in, const int* in_sizes, int n_in,
                              void* d_out, int out_size, void* d_ws, size_t ws_size,
                              hipStream_t stream) {
    const float* z_what      = (const float*)d_in[0];
    const float* z_pres_prob = (const float*)d_in[1];
    const float* z_pres      = (const float*)d_in[2];
    const int*   global_step = (const int*)d_in[3];
    float* out = (float*)d_out;
    float* ws  = (float*)d_ws;

    pair_kernel<<<NPAIR, 256, SMEM_BYTES, stream>>>(z_what, z_pres_prob, ws);
    zpres_kernel<<<ZPRES_BLOCKS, 256, 0, stream>>>(z_pres, ws);
    final_kernel<<<1, 256, 0, stream>>>(ws, global_step, out);
}